// MultiHeadDiffAttention_77833397338628
// MI455X (gfx1250) — compile-verified
//
#include <hip/hip_runtime.h>
#include <hip/hip_bf16.h>

typedef float v2f __attribute__((ext_vector_type(2)));
typedef float v4f __attribute__((ext_vector_type(4)));
typedef float v8f __attribute__((ext_vector_type(8)));

#define BATCH 8
#define NSEQ 1024
#define EMB 512
#define NHEAD 8
#define D2 64            // 2*head_dim
#define NU 6
#define LAMBDA_INIT 0.63212055882855767841f

#define OUT_ELEMS   (BATCH * NSEQ * EMB)            // 4194304
#define ATTNW_ELEMS (BATCH * NHEAD * NSEQ * NSEQ)   // 67108864

// ---------------------------------------------------------------------------
// Kernel 0: beta = sigmoid(exp(dot(lq,lk)) * lambda_init)
// ---------------------------------------------------------------------------
__global__ void beta_kernel(const float* __restrict__ lq, const float* __restrict__ lk,
                            float* __restrict__ wsbeta, float* __restrict__ beta_out) {
    __shared__ float s[128];
    int t = threadIdx.x;
    float p = 0.f;
    for (int i = t; i < 100; i += 128) p += lq[i] * lk[i];
    s[t] = p;
    __syncthreads();
    for (int off = 64; off; off >>= 1) {
        if (t < off) s[t] += s[t + off];
        __syncthreads();
    }
    if (t == 0) {
        float lam = __expf(s[0]);
        float b = 1.f / (1.f + __expf(-lam * LAMBDA_INIT));
        wsbeta[0] = b;
        beta_out[0] = b;
    }
}

// ---------------------------------------------------------------------------
// Kernel 1/3: C[row,f] = sum_e A[row,e] * W[f,e]   (M=8192, N=512, K=512)
// mode 0: store as v layout [B,H,m,64];  mode 1: plain [row,512] (NT store --
// final output, never re-read; keep L2 for the big attention pass)
// fp32 WMMA 16x16x4; A panel staged in LDS (stride 516 -> banks 4m+k, no
// conflicts for the lane0-15 / lane16-31 fragment halves)
// ---------------------------------------------------------------------------
__global__ __launch_bounds__(256) void gemm16_wmma(const float* __restrict__ A,
                                                   const float* __restrict__ W,
                                                   float* __restrict__ Out, int mode) {
    __shared__ float Alds[16 * 516];
    const int tid = threadIdx.x;
    const int mt = blockIdx.x;                 // 16-row tile of 8192 rows

    // stage A tile 16x512 (2048 float4 / 256 threads = 8 each)
    for (int i = tid; i < 2048; i += 256) {
        int r = i >> 7;
        int c4 = (i & 127) << 2;
        *(v4f*)&Alds[r * 516 + c4] =
            *(const v4f*)&A[(size_t)(mt * 16 + r) * 512 + c4];
    }
    __syncthreads();

    const int wave = tid >> 5, lane = tid & 31;
    const int mrow = lane & 15;
    const int k2 = (lane >> 4) << 1;           // 0 for lanes 0-15, 2 for 16-31
    const int mhi = (lane >> 4) << 3;

    for (int t = 0; t < 4; ++t) {
        const int nt = wave * 4 + t;           // N-tile 0..31
        const int n = nt * 16 + (lane & 15);   // output column (f)
        v8f acc = {0.f, 0.f, 0.f, 0.f, 0.f, 0.f, 0.f, 0.f};
        for (int k = 0; k < 512; k += 4) {
            const int kk = k + k2;
            v2f a, b;
            a.x = Alds[mrow * 516 + kk];
            a.y = Alds[mrow * 516 + kk + 1];
            b.x = W[(size_t)n * 512 + kk];     // B[k][f] = W[f*512+k]
            b.y = W[(size_t)n * 512 + kk + 1];
            acc = __builtin_amdgcn_wmma_f32_16x16x4_f32(
                false, a, false, b, (short)0, acc, false, false);
        }
#pragma unroll
        for (int j = 0; j < 8; ++j) {
            const int M = j + mhi;
            const int row = mt * 16 + M;
            if (mode == 0) {
                const int bb = row >> 10, m = row & 1023;
                const int f = nt * 16 + (lane & 15);
                const int h = f >> 6, d = f & 63;
                Out[(((size_t)(bb * NHEAD + h)) * NSEQ + m) * D2 + d] = acc[j];
            } else {
                __builtin_nontemporal_store(
                    acc[j], &Out[(size_t)row * 512 + nt * 16 + (lane & 15)]);
            }
        }
    }
}

// ---------------------------------------------------------------------------
// Kernel 2: fused edge-score projection + softmax + differential combine +
//           attn_w write + attn_w @ v  (one block = (b, h, 16-row tile))
// Two passes over u. Pass 1 fills L2 (u is 201MB vs 192MB L2); pass 2 uses
// NT (last-use) loads. attn_w (268MB, final output, never re-read) is written
// with NT stores so it does not evict u from L2 between the passes.
// ---------------------------------------------------------------------------
#define ASTR 1028   // attn LDS row stride (floats): bank = 4*m + k, conflict-free

__global__ __launch_bounds__(256) void diffattn_kernel(
    const float* __restrict__ u, const float* __restrict__ Wu,
    const float* __restrict__ bu, const float* __restrict__ v,
    const float* __restrict__ betap, float* __restrict__ attnw_out,
    float* __restrict__ attn_out) {
    __shared__ float attnLds[16 * ASTR];
    __shared__ float red[4 * 8 * 32];

    const int tid = threadIdx.x, wave = tid >> 5, lane = tid & 31;
    const int id = blockIdx.x;
    const int rb = id & 63;                 // row tile
    const int h = (id >> 6) & 7;
    const int b = id >> 9;
    const float beta = betap[0];

    // channels for this head: reshape(B,H,2,N,N) -> c0 = 2h, c1 = 2h+1
    const int c0 = 2 * h, c1 = 2 * h + 1;
    float wu0[NU], wu1[NU];
#pragma unroll
    for (int uu = 0; uu < NU; ++uu) {
        wu0[uu] = Wu[c0 * NU + uu];
        wu1[uu] = Wu[c1 * NU + uu];
    }
    const float bb0 = bu[c0], bb1 = bu[c1];
    const size_t ustride = (size_t)NSEQ * NSEQ;  // plane stride per u-channel

    for (int rr = 0; rr < 2; ++rr) {
        const int r = wave * 2 + rr;
        const int n = rb * 16 + r;
        const float* urow = u + (size_t)b * NU * ustride + (size_t)n * NSEQ;

        // pass 1: sum of exp(score) per channel (scores ~N(0,1): exp is safe;
        // exp(s)/sum exp(s) == softmax(s) exactly)
        float sm0 = 0.f, sm1 = 0.f;
        for (int mb = lane * 4; mb < NSEQ; mb += 128) {
            v4f uv[NU];
#pragma unroll
            for (int uu = 0; uu < NU; ++uu)
                uv[uu] = *(const v4f*)(urow + (size_t)uu * ustride + mb);
#pragma unroll
            for (int j = 0; j < 4; ++j) {
                float s0 = bb0, s1 = bb1;
#pragma unroll
                for (int uu = 0; uu < NU; ++uu) {
                    const float t = uv[uu][j];
                    s0 += wu0[uu] * t;
                    s1 += wu1[uu] * t;
                }
                sm0 += __expf(s0);
                sm1 += __expf(s1);
            }
        }
        for (int off = 16; off; off >>= 1) {
            sm0 += __shfl_xor(sm0, off, 32);
            sm1 += __shfl_xor(sm1, off, 32);
        }
        const float inv0 = 1.f / sm0;
        const float inv1 = beta / sm1;

        // pass 2: attn_w = softmax0 - beta*softmax1 -> LDS + global (NT) output
        float* ldsrow = &attnLds[r * ASTR];
        float* gout = attnw_out + ((size_t)(b * NHEAD + h) * NSEQ + n) * NSEQ;
        for (int mb = lane * 4; mb < NSEQ; mb += 128) {
            v4f uv[NU];
#pragma unroll
            for (int uu = 0; uu < NU; ++uu)
                uv[uu] = __builtin_nontemporal_load(
                    (const v4f*)(urow + (size_t)uu * ustride + mb));
            v4f wv;
#pragma unroll
            for (int j = 0; j < 4; ++j) {
                float s0 = bb0, s1 = bb1;
#pragma unroll
                for (int uu = 0; uu < NU; ++uu) {
                    const float t = uv[uu][j];
                    s0 += wu0[uu] * t;
                    s1 += wu1[uu] * t;
                }
                wv[j] = __expf(s0) * inv0 - __expf(s1) * inv1;
            }
            *(v4f*)(ldsrow + mb) = wv;
            __builtin_nontemporal_store(wv, (v4f*)(gout + mb));
        }
    }
    __syncthreads();

    // attn_w[16,1024] @ v[b,h][1024,64] via fp32 WMMA 16x16x4
    // 8 waves: nt = wave&3 (16-col tile), kh = wave>>2 (512-wide K half)
    const int nt = wave & 3, kh = wave >> 2;
    const float* vb = v + (size_t)(b * NHEAD + h) * NSEQ * D2;
    const int col = nt * 16 + (lane & 15);
    const int mrow = lane & 15;
    const int k2 = (lane >> 4) << 1;
    v8f acc = {0.f, 0.f, 0.f, 0.f, 0.f, 0.f, 0.f, 0.f};
    for (int k = kh * 512; k < kh * 512 + 512; k += 4) {
        const int kk = k + k2;
        v2f a, bf;
        a.x = attnLds[mrow * ASTR + kk];
        a.y = attnLds[mrow * ASTR + kk + 1];
        bf.x = vb[(size_t)kk * D2 + col];
        bf.y = vb[(size_t)(kk + 1) * D2 + col];
        acc = __builtin_amdgcn_wmma_f32_16x16x4_f32(
            false, a, false, bf, (short)0, acc, false, false);
    }
    if (kh == 1) {
#pragma unroll
        for (int j = 0; j < 8; ++j) red[(nt * 8 + j) * 32 + lane] = acc[j];
    }
    __syncthreads();
    if (kh == 0) {
        const int mhi = (lane >> 4) << 3;
#pragma unroll
        for (int j = 0; j < 8; ++j) {
            acc[j] += red[(nt * 8 + j) * 32 + lane];
            const int nr = rb * 16 + (j + mhi);
            attn_out[((size_t)b * NSEQ + nr) * EMB + h * D2 + nt * 16 + (lane & 15)] =
                acc[j];
        }
    }
}

// ---------------------------------------------------------------------------
extern "C" void kernel_launch(void* const* d_in, const int* in_sizes, int n_in,
                              void* d_out, int out_size, void* d_ws, size_t ws_size,
                              hipStream_t stream) {
    const float* x  = (const float*)d_in[0];
    const float* u  = (const float*)d_in[1];
    const float* Wv = (const float*)d_in[2];
    const float* Wo = (const float*)d_in[3];
    const float* lq = (const float*)d_in[4];
    const float* lk = (const float*)d_in[5];
    const float* Wu = (const float*)d_in[6];
    const float* bu = (const float*)d_in[7];

    float* out      = (float*)d_out;                 // [B,N,E]
    float* attnw    = out + OUT_ELEMS;               // [B,H,N,N]
    float* beta_out = out + OUT_ELEMS + ATTNW_ELEMS; // scalar

    float* ws     = (float*)d_ws;
    float* wsbeta = ws;                  // 64 floats (padding)
    float* v_ws   = ws + 64;             // [B,H,N,64]  (4194304 floats)
    float* ao_ws  = v_ws + OUT_ELEMS;    // [B,N,E]     (4194304 floats)

    beta_kernel<<<1, 128, 0, stream>>>(lq, lk, wsbeta, beta_out);
    gemm16_wmma<<<512, 256, 0, stream>>>(x, Wv, v_ws, 0);
    diffattn_kernel<<<4096, 256, 0, stream>>>(u, Wu, bu, v_ws, wsbeta, attnw, ao_ws);
    gemm16_wmma<<<512, 256, 0, stream>>>(ao_ws, Wo, out, 1);
}